// PrototypeLayer_66486093742277
// MI455X (gfx1250) — compile-verified
//
#include <hip/hip_runtime.h>

typedef __attribute__((ext_vector_type(2))) float v2f;
typedef __attribute__((ext_vector_type(4))) float v4f;
typedef __attribute__((ext_vector_type(8))) float v8f;
typedef __attribute__((ext_vector_type(4))) int   v4i;

// async-copy pointer types: pointee carries the address space
typedef __attribute__((address_space(1))) v4i* gbl_v4i_ptr;  // global
typedef __attribute__((address_space(3))) v4i* lds_v4i_ptr;  // LDS

#define D_DIM  1024                // contraction length K
#define SUB_D  16                  // subspace dim d (tile edge)
#define KC     32                  // D-chunk per LDS stage
#define NCHUNK (D_DIM / KC)        // 32
#define CH_FLOATS (KC * SUB_D)     // 512 floats per (sample|proto) chunk
#define WAVES  8
#define SPB    4                   // samples per block (register-block rows)
#define PPW    2                   // prototypes per wave (register-block cols)
#define PPB    (WAVES * PPW)       // 16 prototypes per block

#if defined(__has_builtin)
#  if __has_builtin(__builtin_amdgcn_global_load_async_to_lds_b128) && \
      __has_builtin(__builtin_amdgcn_s_wait_asynccnt)
#    define USE_ASYNC_LDS 1
#  endif
#endif
#ifndef USE_ASYNC_LDS
#  define USE_ASYNC_LDS 0
#endif

// 16-byte global -> LDS copy: async DMA path on gfx1250 when available.
__device__ __forceinline__ void copy16_to_lds(const float* g, float* l) {
#if USE_ASYNC_LDS
    __builtin_amdgcn_global_load_async_to_lds_b128(
        (gbl_v4i_ptr)(void*)(g),
        (lds_v4i_ptr)(void*)(l),
        0, 0);
#else
    *(v4f*)l = *(const v4f*)g;
#endif
}

__device__ __forceinline__ void wait_staging() {
#if USE_ASYNC_LDS
    __builtin_amdgcn_s_wait_asynccnt(0);
#endif
    __syncthreads();
}

// Block: 4 samples x 16 prototypes. Wave w owns the 4x2 tile grid
// (b0..b0+3) x (p0+2w, p0+2w+1); each tile is M[b,p] = X_b^T W_p computed with
// V_WMMA_F32_16X16X4_F32 over K=1024 (8 WMMAs per 6 operand loads).
// dist = sum(rel) - mean(rel)*||M||_F^2 (exact for uniform relevances, since
// sum_j s_j^2 = ||M||_F^2 -> the batched 16x16 SVDs vanish algebraically).
__global__ __launch_bounds__(256)
void grassmann_chordal_kernel(const float* __restrict__ xs,
                              const float* __restrict__ xprotos,
                              const float* __restrict__ relevances,
                              float* __restrict__ dists,
                              int P)
{
    __shared__ float As[2][SPB][CH_FLOATS];   // 16 KB, double buffered
    __shared__ float Bs[2][PPB][CH_FLOATS];   // 64 KB, double buffered

    const int tid  = threadIdx.x;
    const int lane = tid & 31;
    const int wave = tid >> 5;
    const int b0   = blockIdx.y * SPB;
    const int pw   = blockIdx.x * PPB + wave * PPW;

    // WMMA f32 16x16x4 operand layout: lane L supplies K = 2*(L>>4)+{0,1},
    // M/N = L&15. In a 64-float block of 4 D-rows x 16, element (k,m) is at
    // k*16+m:
    const int sel = ((lane >> 4) << 5) + (lane & 15);

    v8f acc[SPB][PPW];
    #pragma unroll
    for (int s = 0; s < SPB; ++s)
        #pragma unroll
        for (int q = 0; q < PPW; ++q)
            acc[s][q] = v8f{0.f, 0.f, 0.f, 0.f, 0.f, 0.f, 0.f, 0.f};

    // ---- staging: A cooperatively (512 x 16 B chunks), B per wave ----
    auto stage = [&](int buf, int k0) {
        #pragma unroll
        for (int i = 0; i < 2; ++i) {
            const int c   = tid + i * 256;       // 16-byte chunk id, 0..511
            const int s   = c >> 7;              // which sample
            const int off = (c & 127) * 4;       // float offset within chunk
            copy16_to_lds(
                xs + (size_t)(b0 + s) * (D_DIM * SUB_D) + k0 * SUB_D + off,
                &As[buf][s][off]);
        }
        #pragma unroll
        for (int q = 0; q < PPW; ++q) {
            const float* wgp =
                xprotos + (size_t)(pw + q) * (D_DIM * SUB_D) + k0 * SUB_D;
            #pragma unroll
            for (int i = 0; i < 4; ++i) {
                const int boff = (i * 32 + lane) * 4;
                copy16_to_lds(wgp + boff, &Bs[buf][wave * PPW + q][boff]);
            }
        }
    };

    auto compute = [&](int buf) {
        const float* B0 = Bs[buf][wave * PPW + 0];
        const float* B1 = Bs[buf][wave * PPW + 1];
        #pragma unroll
        for (int step = 0; step < KC / 4; ++step) {
            const int o = step * 64 + sel;
            v2f a[SPB], c0, c1;
            #pragma unroll
            for (int s = 0; s < SPB; ++s) {
                a[s].x = As[buf][s][o];
                a[s].y = As[buf][s][o + 16];
            }
            c0.x = B0[o];  c0.y = B0[o + 16];
            c1.x = B1[o];  c1.y = B1[o + 16];
            #pragma unroll
            for (int s = 0; s < SPB; ++s)
                acc[s][0] = __builtin_amdgcn_wmma_f32_16x16x4_f32(
                    false, a[s], false, c0, (short)0, acc[s][0], false, false);
            #pragma unroll
            for (int s = 0; s < SPB; ++s)
                acc[s][1] = __builtin_amdgcn_wmma_f32_16x16x4_f32(
                    false, a[s], false, c1, (short)0, acc[s][1], false, false);
        }
    };

    // ---- double-buffered main loop: stage c+1 while computing c ----
    stage(0, 0);
    wait_staging();
    for (int c = 0; c < NCHUNK; ++c) {
        const int buf = c & 1;
        if (c + 1 < NCHUNK) stage(buf ^ 1, (c + 1) * KC);
        compute(buf);
        wait_staging();   // staging of c+1 done + all waves done reading buf
    }

    // ---- relevance sum (uniform -> closed form needs only the sum) ----
    float r = (lane < SUB_D) ? relevances[lane] : 0.f;
    #pragma unroll
    for (int off = 8; off > 0; off >>= 1)
        r += __shfl_xor(r, off, 32);           // lane 0: sum over j
    const float sum_rel = r;

    // ---- ||M||_F^2 per tile: per-lane squares + wave32 butterfly ----
    #pragma unroll
    for (int s = 0; s < SPB; ++s) {
        #pragma unroll
        for (int q = 0; q < PPW; ++q) {
            float frob = 0.f;
            #pragma unroll
            for (int i = 0; i < 8; ++i) {
                const float c = acc[s][q][i];
                frob += c * c;
            }
            #pragma unroll
            for (int off = 16; off > 0; off >>= 1)
                frob += __shfl_xor(frob, off, 32);
            if (lane == 0)
                dists[(size_t)(b0 + s) * P + (pw + q)] =
                    sum_rel - sum_rel * (1.0f / SUB_D) * frob;
        }
    }
}

extern "C" void kernel_launch(void* const* d_in, const int* in_sizes, int n_in,
                              void* d_out, int out_size, void* d_ws, size_t ws_size,
                              hipStream_t stream)
{
    const float* xs        = (const float*)d_in[0];  // [B, D, d] fp32
    const float* xprotos   = (const float*)d_in[1];  // [P, D, d] fp32
    const float* relevance = (const float*)d_in[2];  // [1, d]    fp32
    float*       dists     = (float*)d_out;          // [B, P]    fp32

    const int B = in_sizes[0] / (D_DIM * SUB_D);
    const int P = in_sizes[1] / (D_DIM * SUB_D);

    dim3 grid(P / PPB, B / SPB);
    dim3 block(256);
    hipLaunchKernelGGL(grassmann_chordal_kernel, grid, block, 0, stream,
                       xs, xprotos, relevance, dists, P);
}